// get_model_84146999263524
// MI455X (gfx1250) — compile-verified
//
#include <hip/hip_runtime.h>
#include <hip/hip_bf16.h>

// ---------------------------------------------------------------------------
// PointNet++ multi-task seg forward for MI455X (gfx1250), f32 WMMA GEMM core.
// All einsum layers lower to v_wmma_f32_16x16x4_f32 (16x16 tile / wave32).
// GEMM inner loop is guard-free: Cin padded to %4, W padded to full 16-row
// tiles, so A/B operands are plain global_load_b64 feeding the WMMA.
// ---------------------------------------------------------------------------

typedef float v2f __attribute__((ext_vector_type(2)));
typedef float v8f __attribute__((ext_vector_type(8)));

constexpr int BB   = 16;    // batch
constexpr int NPTS = 8192;
constexpr int NCLS = 20;
constexpr int EMBC = 64;
constexpr float EPS = 1e-5f;

// ---------------- transpose input (B,5,N) -> l0_xyz (B,N,3), l0_pts (B,N,5)
__global__ void prep_points(const float* __restrict__ xin,
                            float* __restrict__ l0x, float* __restrict__ l0p) {
  int t = blockIdx.x * blockDim.x + threadIdx.x;
  if (t >= BB * NPTS) return;
  int b = t / NPTS, n = t % NPTS;
  const float* s = xin + (size_t)b * 5 * NPTS;
  float v0 = s[0*NPTS+n], v1 = s[1*NPTS+n], v2 = s[2*NPTS+n];
  float v3 = s[3*NPTS+n], v4 = s[4*NPTS+n];
  float* px = l0x + (size_t)t * 3; px[0]=v0; px[1]=v1; px[2]=v2;
  float* pp = l0p + (size_t)t * 5; pp[0]=v0; pp[1]=v1; pp[2]=v2; pp[3]=v3; pp[4]=v4;
}

// ---------------- farthest point sampling: 1 block per batch, dist[] in LDS
__global__ void fps_kernel(const float* __restrict__ xyz, int* __restrict__ outIdx,
                           int N, int npoint) {
  extern __shared__ float dist[];
  __shared__ float rb[256];
  __shared__ int   ri[256];
  int b = blockIdx.x;
  const float* X = xyz + (size_t)b * N * 3;
  for (int i = threadIdx.x; i < N; i += blockDim.x) dist[i] = 1e10f;
  __syncthreads();
  int far = 0;
  for (int it = 0; it < npoint; ++it) {
    if (threadIdx.x == 0) outIdx[b * npoint + it] = far;
    float cx = X[far*3+0], cy = X[far*3+1], cz = X[far*3+2];
    float bd = -1.0f; int bi = 0;
    for (int i = threadIdx.x; i < N; i += blockDim.x) {
      float dx = X[i*3+0]-cx, dy = X[i*3+1]-cy, dz = X[i*3+2]-cz;
      float d = dx*dx + dy*dy + dz*dz;
      float nd = fminf(dist[i], d);
      dist[i] = nd;
      if (nd > bd) { bd = nd; bi = i; }   // ascending i => first-max kept
    }
    rb[threadIdx.x] = bd; ri[threadIdx.x] = bi;
    __syncthreads();
    for (int o = blockDim.x >> 1; o > 0; o >>= 1) {
      if (threadIdx.x < o) {
        if (rb[threadIdx.x+o] > rb[threadIdx.x] ||
            (rb[threadIdx.x+o] == rb[threadIdx.x] && ri[threadIdx.x+o] < ri[threadIdx.x])) {
          rb[threadIdx.x] = rb[threadIdx.x+o]; ri[threadIdx.x] = ri[threadIdx.x+o];
        }
      }
      __syncthreads();
    }
    far = ri[0];
    __syncthreads();
  }
}

__global__ void gather_xyz(const float* __restrict__ xyz, const int* __restrict__ idx,
                           float* __restrict__ out, int N, int S) {
  int t = blockIdx.x * blockDim.x + threadIdx.x;
  if (t >= BB * S) return;
  int b = t / S;
  const float* p = xyz + ((size_t)b * N + idx[t]) * 3;
  float* o = out + (size_t)t * 3;
  o[0]=p[0]; o[1]=p[1]; o[2]=p[2];
}

// first `ns` in-radius indices in ascending order, padded with the first hit
__global__ void ball_query(const float* __restrict__ xyz, const float* __restrict__ nxyz,
                           int* __restrict__ gidx, int N, int S, int ns, float r2) {
  int t = blockIdx.x * blockDim.x + threadIdx.x;
  if (t >= BB * S) return;
  int b = t / S;
  const float* X = xyz + (size_t)b * N * 3;
  float cx = nxyz[t*3+0], cy = nxyz[t*3+1], cz = nxyz[t*3+2];
  int* o = gidx + (size_t)t * ns;
  int cnt = 0, first = N - 1;
  for (int i = 0; i < N && cnt < ns; ++i) {
    float dx = X[i*3]-cx, dy = X[i*3+1]-cy, dz = X[i*3+2]-cz;
    if (dx*dx + dy*dy + dz*dz <= r2) {
      if (cnt == 0) first = i;
      o[cnt++] = i;
    }
  }
  for (; cnt < ns; ++cnt) o[cnt] = first;
}

// grouped features: [ (xyz[gi]-center)(3) | pts[gi](Cp) | 0-pad ] rows of Cpad
__global__ void group_kernel(const float* __restrict__ xyz, const float* __restrict__ pts,
                             const float* __restrict__ nxyz, const int* __restrict__ gidx,
                             float* __restrict__ out, int N, int S, int ns, int Cp, int Cpad) {
  int t = blockIdx.x * blockDim.x + threadIdx.x;
  if (t >= BB * S * ns) return;
  int ps = t / ns; int b = ps / S;
  int gi = gidx[t];
  const float* xp = xyz + ((size_t)b * N + gi) * 3;
  const float* cp = nxyz + (size_t)ps * 3;
  float* o = out + (size_t)t * Cpad;
  o[0]=xp[0]-cp[0]; o[1]=xp[1]-cp[1]; o[2]=xp[2]-cp[2];
  const float* fp = pts + ((size_t)b * N + gi) * Cp;
  for (int c = 0; c < Cp; ++c) o[3+c] = fp[c];
  for (int c = 3 + Cp; c < Cpad; ++c) o[c] = 0.f;
}

// zero-padded weight copy: o[r][c] (rowsPad x cinPad) from w (rows x cin)
__global__ void pad_weight(const float* __restrict__ w, float* __restrict__ o,
                           int rows, int cin, int rowsPad, int cinPad) {
  int t = blockIdx.x * blockDim.x + threadIdx.x;
  if (t >= rowsPad * cinPad) return;
  int c = t % cinPad, r = t / cinPad;
  o[t] = (r < rows && c < cin) ? w[(size_t)r * cin + c] : 0.f;
}

// ---------------- WMMA GEMM: Y[P][Cout] = A[P][Cin] * W[CoutPad][ldw]^T + bias
// Requirements: Cin % 4 == 0, CoutPad % 16 == 0, W zero-padded to CoutPad rows.
// One 16x16 output tile per wave; K stepped by 4 via v_wmma_f32_16x16x4_f32.
// Inner loop: two global_load_b64 + one WMMA, no predication (EXEC all-1s).
// rowsPerBatch>0 => bias is per-batch [B][Cout], batch = row / rowsPerBatch.
__global__ void gemm_wmma(const float* __restrict__ A, const float* __restrict__ W, int ldw,
                          const float* __restrict__ bias, int rowsPerBatch,
                          float* __restrict__ Y, int P, int Cin, int Cout, int CoutPad) {
  int wave = threadIdx.x >> 5;
  int lane = threadIdx.x & 31;
  int tilesN = CoutPad >> 4;
  int nTiles = (P >> 4) * tilesN;
  int tileId = blockIdx.x * 4 + wave;
  if (tileId >= nTiles) return;           // wave-uniform exit: EXEC stays all-1s
  int tm = (tileId / tilesN) << 4;
  int tn = (tileId % tilesN) << 4;
  int half = lane >> 4, l16 = lane & 15;
  // A: lane = M row; lanes<16 carry K{k,k+1}, lanes>=16 carry K{k+2,k+3}
  const float* Ar = A + (size_t)(tm + l16) * Cin + 2 * half;
  // B: lane = N (output channel); same K split
  const float* Wr = W + (size_t)(tn + l16) * ldw + 2 * half;
  v8f acc = {};
#pragma unroll 4
  for (int k = 0; k < Cin; k += 4) {
    v2f a  = *(const v2f*)(Ar + k);
    v2f bm = *(const v2f*)(Wr + k);
    acc = __builtin_amdgcn_wmma_f32_16x16x4_f32(false, a, false, bm,
                                                (short)0, acc, false, false);
  }
  int col = tn + l16;
  if (col < Cout) {                        // only head L2 (20 of 32) masks here
    for (int r = 0; r < 8; ++r) {
      int row = tm + r + 8 * half;         // C/D layout: M = r + 8*half
      float bv = (rowsPerBatch > 0) ? bias[(row / rowsPerBatch) * Cout + col]
                                    : bias[col];
      Y[(size_t)row * Cout + col] = acc[r] + bv;
    }
  }
}

// ---------------- global batch-norm: per-channel mean / var over all rows
__global__ void bn_stats(const float* __restrict__ Y, float* __restrict__ stats,
                         int P, int C) {
  __shared__ float sh[256], sh2[256];
  int c = blockIdx.x;
  float s = 0.f, s2 = 0.f;
  for (int p = threadIdx.x; p < P; p += blockDim.x) {
    float v = Y[(size_t)p * C + c];
    s += v; s2 += v * v;
  }
  sh[threadIdx.x] = s; sh2[threadIdx.x] = s2;
  __syncthreads();
  for (int o = blockDim.x >> 1; o > 0; o >>= 1) {
    if (threadIdx.x < o) { sh[threadIdx.x] += sh[threadIdx.x+o]; sh2[threadIdx.x] += sh2[threadIdx.x+o]; }
    __syncthreads();
  }
  if (threadIdx.x == 0) {
    float m = sh[0] / (float)P;
    stats[c] = m;
    stats[C + c] = sh2[0] / (float)P - m * m;   // population variance (ddof=0)
  }
}

__global__ void bn_apply(const float* __restrict__ Y, const float* __restrict__ stats,
                         const float* __restrict__ g, const float* __restrict__ be,
                         float* __restrict__ Out, int total, int C) {
  int i = blockIdx.x * blockDim.x + threadIdx.x;
  if (i >= total) return;
  int c = i % C;
  float x = (Y[i] - stats[c]) * rsqrtf(stats[C + c] + EPS) * g[c] + be[c];
  Out[i] = x > 0.f ? x : 0.f;
}

__global__ void maxpool_ns(const float* __restrict__ act, float* __restrict__ out,
                           int PS, int ns, int C) {
  int t = blockIdx.x * blockDim.x + threadIdx.x;
  if (t >= PS * C) return;
  int c = t % C, p = t / C;
  const float* a = act + (size_t)p * ns * C + c;
  float m = a[0];
  for (int j = 1; j < ns; ++j) m = fmaxf(m, a[(size_t)j * C]);
  out[t] = m;
}

// ---------------- 3-NN (sorted ascending dist, first-index tie-break) + weights
__global__ void knn3(const float* __restrict__ x1, const float* __restrict__ x2,
                     int* __restrict__ idx3, float* __restrict__ w3, int N1, int N2) {
  int t = blockIdx.x * blockDim.x + threadIdx.x;
  if (t >= BB * N1) return;
  int b = t / N1;
  float cx = x1[t*3+0], cy = x1[t*3+1], cz = x1[t*3+2];
  const float* X2 = x2 + (size_t)b * N2 * 3;
  float d0 = 3.4e38f, d1 = 3.4e38f, d2 = 3.4e38f;
  int   i0 = 0, i1 = 0, i2 = 0;
  for (int i = 0; i < N2; ++i) {
    float dx = X2[i*3]-cx, dy = X2[i*3+1]-cy, dz = X2[i*3+2]-cz;
    float d = dx*dx + dy*dy + dz*dz;
    if (d < d0)      { d2=d1; i2=i1; d1=d0; i1=i0; d0=d; i0=i; }
    else if (d < d1) { d2=d1; i2=i1; d1=d;  i1=i; }
    else if (d < d2) { d2=d;  i2=i; }
  }
  float w0 = 1.f/(d0+1e-8f), w1 = 1.f/(d1+1e-8f), w2 = 1.f/(d2+1e-8f);
  float s = w0 + w1 + w2;
  idx3[t*3+0]=i0; idx3[t*3+1]=i1; idx3[t*3+2]=i2;
  w3[t*3+0]=w0/s; w3[t*3+1]=w1/s; w3[t*3+2]=w2/s;
}

__global__ void interp3(const float* __restrict__ pts2, const int* __restrict__ idx3,
                        const float* __restrict__ w3, float* __restrict__ out,
                        int N1, int N2, int C) {
  int t = blockIdx.x * blockDim.x + threadIdx.x;
  int total = BB * N1 * C;
  if (t >= total) return;
  int c = t % C, p = t / C, b = p / N1;
  const float* base = pts2 + (size_t)b * N2 * C;
  float v = 0.f;
  for (int k = 0; k < 3; ++k)
    v += w3[p*3+k] * base[(size_t)idx3[p*3+k] * C + c];
  out[t] = v;
}

// fp2 feats concat: [ l1_points(64) | interp(l2,128) | emb[task](64) ]
__global__ void fp2_feats(const float* __restrict__ l1p, const float* __restrict__ l2p,
                          const int* __restrict__ idx3, const float* __restrict__ w3,
                          const float* __restrict__ emb, const int* __restrict__ task,
                          float* __restrict__ feats) {
  int t = blockIdx.x * blockDim.x + threadIdx.x;
  if (t >= BB * 128 * 256) return;
  int c = t % 256, p = t / 256, b = p / 128;
  float v;
  if (c < 64) v = l1p[(size_t)p * 64 + c];
  else if (c < 192) {
    int cc = c - 64; v = 0.f;
    const float* base = l2p + (size_t)b * 32 * 128;
    for (int k = 0; k < 3; ++k)
      v += w3[p*3+k] * base[(size_t)idx3[p*3+k] * 128 + cc];
  } else v = emb[(size_t)task[b] * EMBC + (c - 192)];
  feats[t] = v;
}

// fp1 layer0: fold W[:,128:192] @ emb[task] into a per-batch effective bias
__global__ void fp1_bias(const float* __restrict__ w, const float* __restrict__ bvec,
                         const float* __restrict__ emb, const int* __restrict__ task,
                         float* __restrict__ be) {
  int t = blockIdx.x * blockDim.x + threadIdx.x;
  if (t >= BB * 128) return;
  int b = t / 128, o = t % 128;
  const float* e = emb + (size_t)task[b] * EMBC;
  const float* wr = w + (size_t)o * 192 + 128;
  float s = bvec[o];
  for (int k = 0; k < EMBC; ++k) s += wr[k] * e[k];
  be[t] = s;
}

__global__ void log_softmax20(float* __restrict__ x, int P) {
  int p = blockIdx.x * blockDim.x + threadIdx.x;
  if (p >= P) return;
  float* r = x + (size_t)p * NCLS;
  float m = r[0];
  for (int c = 1; c < NCLS; ++c) m = fmaxf(m, r[c]);
  float s = 0.f;
  for (int c = 0; c < NCLS; ++c) s += __expf(r[c] - m);
  float ls = __logf(s) + m;
  for (int c = 0; c < NCLS; ++c) r[c] = r[c] - ls;
}

// out2[b][c][n] = l1_new[b][n][c]
__global__ void transpose_l1(const float* __restrict__ l1n, float* __restrict__ o2) {
  int t = blockIdx.x * blockDim.x + threadIdx.x;
  if (t >= BB * 128 * 128) return;
  int n = t % 128, c = (t / 128) % 128, b = t / (128 * 128);
  o2[t] = l1n[((size_t)b * 128 + n) * 128 + c];
}

// ---------------------------------------------------------------------------
extern "C" void kernel_launch(void* const* d_in, const int* in_sizes, int n_in,
                              void* d_out, int out_size, void* d_ws, size_t ws_size,
                              hipStream_t stream) {
  // Input order: jax pytree flatten, dicts by sorted key.
  // 0 xyz, 1 task_ids, then params: fp1{emb,mlp[3]{b,beta,g,w}}, fp2{emb,mlp[2]},
  // head{b1,b2,beta1,g1,w1,w2}, sa1[3]{b,beta,g,w}, sa2[3]{b,beta,g,w}
  const float* xin  = (const float*)d_in[0];
  const int*   task = (const int*)  d_in[1];
  const float* fp1_emb = (const float*)d_in[2];
  const float *fp1b0=(const float*)d_in[3],  *fp1be0=(const float*)d_in[4],  *fp1g0=(const float*)d_in[5],  *fp1w0=(const float*)d_in[6];
  const float *fp1b1=(const float*)d_in[7],  *fp1be1=(const float*)d_in[8],  *fp1g1=(const float*)d_in[9],  *fp1w1=(const float*)d_in[10];
  const float *fp1b2=(const float*)d_in[11], *fp1be2=(const float*)d_in[12], *fp1g2=(const float*)d_in[13], *fp1w2=(const float*)d_in[14];
  const float* fp2_emb = (const float*)d_in[15];
  const float *fp2b0=(const float*)d_in[16], *fp2be0=(const float*)d_in[17], *fp2g0=(const float*)d_in[18], *fp2w0=(const float*)d_in[19];
  const float *fp2b1=(const float*)d_in[20], *fp2be1=(const float*)d_in[21], *fp2g1=(const float*)d_in[22], *fp2w1=(const float*)d_in[23];
  const float *hb1=(const float*)d_in[24], *hb2=(const float*)d_in[25], *hbe1=(const float*)d_in[26], *hg1=(const float*)d_in[27], *hw1=(const float*)d_in[28], *hw2=(const float*)d_in[29];
  const float *s1b0=(const float*)d_in[30], *s1be0=(const float*)d_in[31], *s1g0=(const float*)d_in[32], *s1w0=(const float*)d_in[33];
  const float *s1b1=(const float*)d_in[34], *s1be1=(const float*)d_in[35], *s1g1=(const float*)d_in[36], *s1w1=(const float*)d_in[37];
  const float *s1b2=(const float*)d_in[38], *s1be2=(const float*)d_in[39], *s1g2=(const float*)d_in[40], *s1w2=(const float*)d_in[41];
  const float *s2b0=(const float*)d_in[42], *s2be0=(const float*)d_in[43], *s2g0=(const float*)d_in[44], *s2w0=(const float*)d_in[45];
  const float *s2b1=(const float*)d_in[46], *s2be1=(const float*)d_in[47], *s2g1=(const float*)d_in[48], *s2w1=(const float*)d_in[49];
  const float *s2b2=(const float*)d_in[50], *s2be2=(const float*)d_in[51], *s2g2=(const float*)d_in[52], *s2w2=(const float*)d_in[53];
  float* out = (float*)d_out;

  // ---- workspace carve (floats) ----
  float* ws = (float*)d_ws;
  size_t off = 0;
  auto alloc = [&](size_t n) { float* p = ws + off; off += n; return p; };
  float* l0x    = alloc((size_t)BB*NPTS*3);
  float* l0p    = alloc((size_t)BB*NPTS*5);
  float* nx1    = alloc((size_t)BB*128*3);
  float* nx2    = alloc((size_t)BB*32*3);
  float* l1pts  = alloc((size_t)BB*128*64);
  float* l2pts  = alloc((size_t)BB*32*128);
  float* l1new  = alloc((size_t)BB*128*128);
  float* feats2 = alloc((size_t)BB*128*256);
  float* f2a    = alloc((size_t)BB*128*256);
  float* f2b    = alloc((size_t)BB*128*128);
  float* beff   = alloc((size_t)BB*128);
  float* stats  = alloc(512);
  float* s2w0p  = alloc((size_t)64*68);     // W_sa2_0 padded 67 -> 68 cols
  float* hw2p   = alloc((size_t)32*128);    // W_head_2 padded 20 -> 32 rows
  float* kw     = alloc((size_t)BB*NPTS*3);
  int* fps1  = (int*)alloc((size_t)BB*128);
  int* fps2  = (int*)alloc((size_t)BB*32);
  int* gidx1 = (int*)alloc((size_t)BB*128*32);
  int* gidx2 = (int*)alloc((size_t)BB*32*32);
  int* kidx  = (int*)alloc((size_t)BB*NPTS*3);
  float* bigA = alloc((size_t)BB*NPTS*128);   // 16.7M floats
  float* bigB = alloc((size_t)BB*NPTS*128);
  // SA-stage temporaries alias inside bigA / bigB (disjoint lifetimes)
  float* gbuf1 = bigA;                         // [65536][8]
  float* t1a   = bigA + (1u<<20);              // [65536][<=64]
  float* t1b   = t1a  + (1u<<22);
  float* gbuf2 = bigB;                         // [16384][68]
  float* t2a   = bigB + (1u<<21);              // [16384][<=128]
  float* t2b   = t2a  + (1u<<21);

  auto GEMM = [&](const float* A, const float* W, int ldw, const float* bias,
                  int rpb, float* Y, int P, int Cin, int Cout, int CoutPad) {
    int nTiles = (P >> 4) * (CoutPad >> 4);
    gemm_wmma<<<(nTiles + 3) / 4, 128, 0, stream>>>(A, W, ldw, bias, rpb, Y,
                                                    P, Cin, Cout, CoutPad);
  };
  auto BN = [&](float* Y, int P, int C, const float* g, const float* be, float* Out) {
    bn_stats<<<C, 256, 0, stream>>>(Y, stats, P, C);
    int tot = P * C;
    bn_apply<<<(tot + 255) / 256, 256, 0, stream>>>(Y, stats, g, be, Out, tot, C);
  };

  // ---- weight padding (tiny, once per launch) ----
  pad_weight<<<(64*68 + 255)/256, 256, 0, stream>>>(s2w0, s2w0p, 64, 67, 64, 68);
  pad_weight<<<(32*128 + 255)/256, 256, 0, stream>>>(hw2, hw2p, 20, 128, 32, 128);

  // ---- input transpose ----
  prep_points<<<(BB*NPTS + 255)/256, 256, 0, stream>>>(xin, l0x, l0p);

  // ---- SA1: npoint=128, r=0.1, ns=32, mlp 8->32->32->64 ----
  fps_kernel<<<BB, 256, NPTS*sizeof(float), stream>>>(l0x, fps1, NPTS, 128);
  gather_xyz<<<(BB*128 + 255)/256, 256, 0, stream>>>(l0x, fps1, nx1, NPTS, 128);
  ball_query<<<(BB*128 + 255)/256, 256, 0, stream>>>(l0x, nx1, gidx1, NPTS, 128, 32, 0.01f);
  group_kernel<<<(BB*128*32 + 255)/256, 256, 0, stream>>>(l0x, l0p, nx1, gidx1, gbuf1, NPTS, 128, 32, 5, 8);
  const int P1 = BB*128*32;
  GEMM(gbuf1, s1w0, 8,  s1b0, 0, t1a, P1, 8, 32, 32);   BN(t1a, P1, 32, s1g0, s1be0, t1a);
  GEMM(t1a,   s1w1, 32, s1b1, 0, t1b, P1, 32, 32, 32);  BN(t1b, P1, 32, s1g1, s1be1, t1b);
  GEMM(t1b,   s1w2, 32, s1b2, 0, t1a, P1, 32, 64, 64);  BN(t1a, P1, 64, s1g2, s1be2, t1a);
  maxpool_ns<<<(BB*128*64 + 255)/256, 256, 0, stream>>>(t1a, l1pts, BB*128, 32, 64);

  // ---- SA2: npoint=32, r=0.2, ns=32, mlp 67(->68)->64->64->128 ----
  fps_kernel<<<BB, 256, 128*sizeof(float), stream>>>(nx1, fps2, 128, 32);
  gather_xyz<<<(BB*32 + 255)/256, 256, 0, stream>>>(nx1, fps2, nx2, 128, 32);
  ball_query<<<(BB*32 + 255)/256, 256, 0, stream>>>(nx1, nx2, gidx2, 128, 32, 32, 0.04f);
  group_kernel<<<(BB*32*32 + 255)/256, 256, 0, stream>>>(nx1, l1pts, nx2, gidx2, gbuf2, 128, 32, 32, 64, 68);
  const int P2 = BB*32*32;
  GEMM(gbuf2, s2w0p, 68, s2b0, 0, t2a, P2, 68, 64, 64);  BN(t2a, P2, 64, s2g0, s2be0, t2a);
  GEMM(t2a,   s2w1,  64, s2b1, 0, t2b, P2, 64, 64, 64);  BN(t2b, P2, 64, s2g1, s2be1, t2b);
  GEMM(t2b,   s2w2,  64, s2b2, 0, t2a, P2, 64, 128, 128); BN(t2a, P2, 128, s2g2, s2be2, t2a);
  maxpool_ns<<<(BB*32*128 + 255)/256, 256, 0, stream>>>(t2a, l2pts, BB*32, 32, 128);

  // ---- FP2: l1 <- interp(l2) ; mlp 256->256->128 ----
  knn3<<<(BB*128 + 255)/256, 256, 0, stream>>>(nx1, nx2, kidx, kw, 128, 32);
  fp2_feats<<<(BB*128*256 + 255)/256, 256, 0, stream>>>(l1pts, l2pts, kidx, kw, fp2_emb, task, feats2);
  const int Pf2 = BB*128;
  GEMM(feats2, fp2w0, 256, fp2b0, 0, f2a, Pf2, 256, 256, 256); BN(f2a, Pf2, 256, fp2g0, fp2be0, f2a);
  GEMM(f2a,    fp2w1, 256, fp2b1, 0, f2b, Pf2, 256, 128, 128); BN(f2b, Pf2, 128, fp2g1, fp2be1, l1new);

  // ---- FP1: l0 <- interp(l1new) ; emb folded into per-batch bias; mlp 192->128->128->128
  knn3<<<(BB*NPTS + 255)/256, 256, 0, stream>>>(l0x, nx1, kidx, kw, NPTS, 128);
  interp3<<<(BB*NPTS*128 + 255)/256, 256, 0, stream>>>(l1new, kidx, kw, bigA, NPTS, 128, 128);
  fp1_bias<<<(BB*128 + 255)/256, 256, 0, stream>>>(fp1w0, fp1b0, fp1_emb, task, beff);
  const int P0 = BB*NPTS;
  GEMM(bigA, fp1w0, 192, beff,  NPTS, bigB, P0, 128, 128, 128); BN(bigB, P0, 128, fp1g0, fp1be0, bigB);
  GEMM(bigB, fp1w1, 128, fp1b1, 0,    bigA, P0, 128, 128, 128); BN(bigA, P0, 128, fp1g1, fp1be1, bigA);
  GEMM(bigA, fp1w2, 128, fp1b2, 0,    bigB, P0, 128, 128, 128); BN(bigB, P0, 128, fp1g2, fp1be2, bigB);

  // ---- head: 128->128 bn_relu -> 128->20(pad 32) -> log_softmax ----
  GEMM(bigB, hw1, 128, hb1, 0, bigA, P0, 128, 128, 128); BN(bigA, P0, 128, hg1, hbe1, bigA);
  GEMM(bigA, hw2p, 128, hb2, 0, out, P0, 128, NCLS, 32);
  log_softmax20<<<(P0 + 255)/256, 256, 0, stream>>>(out, P0);

  // ---- second output: transpose(l1new, (0,2,1)) ----
  transpose_l1<<<(BB*128*128 + 255)/256, 256, 0, stream>>>(l1new, out + (size_t)P0 * NCLS);
}